// PartitionPadding_13116830122452
// MI455X (gfx1250) — compile-verified
//
#include <hip/hip_runtime.h>
#include <stdint.h>

// ---------------------------------------------------------------------------
// PartitionPadding for MI455X (gfx1250).
// Pure data movement: ~512 MB read + ~538 MB write -> ~45 us at 23.3 TB/s.
// Strategy: gather formulation (write output exactly once), CDNA5 async
// global->LDS->global streaming path for valid rows, direct nontemporal
// zero stores for padding rows.
// ---------------------------------------------------------------------------

#define BATCH_P     50
#define MAX_ATOMS_P 21000
#define DIM         128
#define TOTAL_ROWS  (BATCH_P * MAX_ATOMS_P)   // 1,050,000 rows of 512 B

#define WAVES_PER_BLOCK   8
#define ROWS_PER_WAVE     8
#define ROWS_PER_BLOCK    (WAVES_PER_BLOCK * ROWS_PER_WAVE)   // 64
#define THREADS_PER_BLOCK (WAVES_PER_BLOCK * 32)              // 256 (8 wave32)

typedef float f32x4 __attribute__((ext_vector_type(4)));
typedef int   v4i   __attribute__((ext_vector_type(4)));

#if defined(__AMDGCN__) && defined(__gfx1250__) && \
    __has_builtin(__builtin_amdgcn_global_load_async_to_lds_b128) && \
    __has_builtin(__builtin_amdgcn_global_store_async_from_lds_b128)
#define USE_ASYNC 1
// Builtin params are int4 pointers in explicit address spaces:
//   arg0: global (AS1) int4*,  arg1: LDS (AS3) int4*,  then imm offset, imm cpol
typedef __attribute__((address_space(1))) v4i* gptr_t;
typedef __attribute__((address_space(3))) v4i* lptr_t;
#else
#define USE_ASYNC 0
#endif

static __device__ __forceinline__ void wait_async_zero() {
#if USE_ASYNC
#if __has_builtin(__builtin_amdgcn_s_wait_asynccnt)
    __builtin_amdgcn_s_wait_asynccnt(0);
#else
    asm volatile("s_wait_asynccnt 0" ::: "memory");
#endif
#endif
}

// Kernel 1: partition boundaries via binary search on the sorted index array.
// bounds[b] = lower_bound(idx, b) for b in [0, BATCH]; bounds[BATCH] == N.
__global__ void partition_bounds_kernel(const int* __restrict__ idx, int N,
                                        int* __restrict__ bounds) {
    int b = threadIdx.x;
    if (b > BATCH_P) return;
    int lo = 0, hi = N;
    while (lo < hi) {
        int mid = (lo + hi) >> 1;
        if (idx[mid] < b) lo = mid + 1; else hi = mid;
    }
    bounds[b] = lo;
}

// Kernel 2: one wave per 8 output rows of 512 B (lane handles 16 B = f32x4).
//   out[b][p][:] = (p < count[b]) ? src[start[b] + p][:] : 0
__global__ __launch_bounds__(THREADS_PER_BLOCK)
void partition_pad_kernel(const float* __restrict__ src,
                          float* __restrict__ out,
                          const int* __restrict__ bounds) {
#if USE_ASYNC
    __shared__ float tile[WAVES_PER_BLOCK][ROWS_PER_WAVE][DIM];  // 32 KB
#endif
    const int wave = threadIdx.x >> 5;
    const int lane = threadIdx.x & 31;
    const int baseRow = blockIdx.x * ROWS_PER_BLOCK + wave * ROWS_PER_WAVE;

    bool copied[ROWS_PER_WAVE];

#pragma unroll
    for (int r = 0; r < ROWS_PER_WAVE; ++r) {
        copied[r] = false;
        const int row = baseRow + r;
        if (row >= TOTAL_ROWS) continue;
        const int b = row / MAX_ATOMS_P;          // wave-uniform
        const int p = row - b * MAX_ATOMS_P;      // wave-uniform
        const int s0  = bounds[b];
        const int cnt = bounds[b + 1] - s0;
        if (p < cnt) {
            const float* g = src + (size_t)(s0 + p) * DIM + lane * 4;
#if USE_ASYNC
            // 16 B per lane, 512 B per wave, ASYNCcnt-tracked, no VGPR data.
            __builtin_amdgcn_global_load_async_to_lds_b128(
                (gptr_t)g,
                (lptr_t)&tile[wave][r][lane * 4],
                0, 0);
            copied[r] = true;
#else
            f32x4 v = __builtin_nontemporal_load((const f32x4*)g);
            __builtin_nontemporal_store(v,
                (f32x4*)(out + (size_t)row * DIM + lane * 4));
#endif
        } else {
            // Padding row: stream zeros straight out (no cache pollution).
            f32x4 z = (f32x4){0.f, 0.f, 0.f, 0.f};
            __builtin_nontemporal_store(z,
                (f32x4*)(out + (size_t)row * DIM + lane * 4));
        }
    }

#if USE_ASYNC
    wait_async_zero();   // all 8 row tiles resident in LDS
#pragma unroll
    for (int r = 0; r < ROWS_PER_WAVE; ++r) {
        const int row = baseRow + r;
        if (row >= TOTAL_ROWS || !copied[r]) continue;
        float* g = out + (size_t)row * DIM + lane * 4;
        __builtin_amdgcn_global_store_async_from_lds_b128(
            (gptr_t)g,
            (lptr_t)&tile[wave][r][lane * 4],
            0, 0);
    }
    wait_async_zero();   // stores issued to memory before wave retires
#endif
}

extern "C" void kernel_launch(void* const* d_in, const int* in_sizes, int n_in,
                              void* d_out, int out_size, void* d_ws, size_t ws_size,
                              hipStream_t stream) {
    const float* atom_features = (const float*)d_in[0];
    const int*   part_idx      = (const int*)d_in[1];   // values in [0, 50)
    const int    N             = in_sizes[1];           // 1,000,000

    int* bounds = (int*)d_ws;                           // 51 ints of scratch

    partition_bounds_kernel<<<1, 64, 0, stream>>>(part_idx, N, bounds);

    const int grid = (TOTAL_ROWS + ROWS_PER_BLOCK - 1) / ROWS_PER_BLOCK; // 16407
    partition_pad_kernel<<<grid, THREADS_PER_BLOCK, 0, stream>>>(
        atom_features, (float*)d_out, bounds);
}